// MyDropout_11484742550204
// MI455X (gfx1250) — compile-verified
//
#include <hip/hip_runtime.h>

// Elementwise dropout-mask-multiply, 4096x4096 f32. Memory-bound:
// 201 MB traffic -> ~8.6 us floor at 23.3 TB/s HBM. One block per row:
// 1024 float4/row = 256 threads x 4 vec4, with k*4KB immediate offsets so
// address math is computed once per buffer. All 8 b128 loads issued as one
// clause, NT b128 stores keep x/u (128 MiB) resident in the 192 MB L2.

typedef float v4f __attribute__((ext_vector_type(4)));

#define DROP_P      0.1f
#define SCALE       (1.0f / (1.0f - DROP_P))
#define NCOLS       4096
#define VEC_PER_ROW (NCOLS / 4)     // 1024 float4 per row
#define CHUNK       256             // threads per block; vec4 stride per chunk
#define NCHUNK      (VEC_PER_ROW / CHUNK)   // 4 chunks per thread
#define FIRST_VECS  (64 / 4)        // first 64 cols = first 16 float4 of chunk 0

__global__ __launch_bounds__(CHUNK) void dropout_row_kernel(
    const v4f* __restrict__ x4,
    const v4f* __restrict__ u4,
    const int* __restrict__ flagp,
    v4f* __restrict__ o4)
{
    const int  flag = flagp[0];                       // uniform -> s_load
    const int  t    = threadIdx.x;
    const long base = (long)blockIdx.x * VEC_PER_ROW + t;

    const v4f* xp = x4 + base;                        // one 64-bit base each;
    const v4f* up = u4 + base;                        // chunks use IOFFSET imm
    v4f*       op = o4 + base;

    // Issue all loads back-to-back (s_clause of global_load_b128)
    v4f xv[NCHUNK], uv[NCHUNK];
#pragma unroll
    for (int k = 0; k < NCHUNK; ++k) {
        xv[k] = xp[k * CHUNK];
        uv[k] = up[k * CHUNK];
    }

#pragma unroll
    for (int k = 0; k < NCHUNK; ++k) {
        v4f r;
        r.x = xv[k].x * ((uv[k].x >= DROP_P) ? SCALE : 0.0f);
        r.y = xv[k].y * ((uv[k].y >= DROP_P) ? SCALE : 0.0f);
        r.z = xv[k].z * ((uv[k].z >= DROP_P) ? SCALE : 0.0f);
        r.w = xv[k].w * ((uv[k].w >= DROP_P) ? SCALE : 0.0f);

        // First-64-column band exists only in chunk 0, lanes 0..15
        if (k == 0 && t < FIRST_VECS) {
            if (flag == 1) {                  // zero cols, then dropout -> 0
                r.x = 0.0f; r.y = 0.0f; r.z = 0.0f; r.w = 0.0f;
            } else {                          // dropout, then overwrite with 2
                r.x = 2.0f * xv[k].x;
                r.y = 2.0f * xv[k].y;
                r.z = 2.0f * xv[k].z;
                r.w = 2.0f * xv[k].w;
            }
        }

        // NT store: out is never read; don't evict x/u from L2.
        __builtin_nontemporal_store(r, &op[k * CHUNK]);
    }
}

extern "C" void kernel_launch(void* const* d_in, const int* in_sizes, int n_in,
                              void* d_out, int out_size, void* d_ws, size_t ws_size,
                              hipStream_t stream) {
    const v4f* x4 = (const v4f*)d_in[0];
    const v4f* u4 = (const v4f*)d_in[1];
    const int* fl = (const int*)d_in[2];
    v4f*       o4 = (v4f*)d_out;

    const int rows = out_size / NCOLS;        // 4096 blocks, one per row
    dropout_row_kernel<<<rows, CHUNK, 0, stream>>>(x4, u4, fl, o4);
}